// HomoGNNModel_80865644249442
// MI455X (gfx1250) — compile-verified
//
#include <hip/hip_runtime.h>
#include <hip/hip_bf16.h>

#define D_HID 128
#define EPS_LN 1e-5f

typedef __attribute__((ext_vector_type(2))) float v2f;
typedef __attribute__((ext_vector_type(8))) float v8f;

// ---------------------------------------------------------------------------
// Degree / normalization
// ---------------------------------------------------------------------------
__global__ void k_fill1(float* __restrict__ p, int n) {
    int i = blockIdx.x * blockDim.x + threadIdx.x;
    if (i < n) p[i] = 1.0f;  // self loop contributes 1 to every node's degree
}

__global__ void k_deg_accum(const int* __restrict__ dst, float* __restrict__ deg, int e) {
    int i = blockIdx.x * blockDim.x + threadIdx.x;
    if (i < e) unsafeAtomicAdd(&deg[dst[i]], 1.0f);
}

__global__ void k_rsqrt_inplace(float* __restrict__ p, int n) {
    int i = blockIdx.x * blockDim.x + threadIdx.x;
    if (i < n) p[i] = rsqrtf(p[i]);  // deg >= 1 always (self loops)
}

// ---------------------------------------------------------------------------
// f32 WMMA GEMM: C[M,NOUT] = A[M,K] @ B[K,NOUT] (+ bias). Row-major.
// One wave computes a full 16-row x NOUT block: TILES_N accumulators, the A
// fragment (float2 per lane) is loaded once per K-step and reused across all
// N-tiles. Out-of-range rows/cols are handled by CLAMPING the load indices
// (garbage only lands in D rows/cols that are never stored) so EXEC stays
// all-ones with zero divergence around v_wmma_f32_16x16x4_f32.
// Fragment layouts per ISA 7.12.2:
//   A 16x4 : m = lane&15, k = (lane>>4)*2 + {0,1}
//   B 4x16 : n = lane&15, k = (lane>>4)*2 + {0,1}
//   C 16x16: row = v + 8*(lane>>4), col = lane&15   (v8f)
// ---------------------------------------------------------------------------
template <int K, int NOUT>
__global__ __launch_bounds__(256) void k_gemm_f32_wmma(
    const float* __restrict__ A, const float* __restrict__ B,
    const float* __restrict__ bias, float* __restrict__ C, int M) {
    constexpr int TILES_N = (NOUT + 15) / 16;

    int waveId = (blockIdx.x * blockDim.x + threadIdx.x) >> 5;
    int lane   = threadIdx.x & 31;
    int tilesM = (M + 15) >> 4;
    if (waveId >= tilesM) return;  // wave-uniform -> EXEC stays all-ones

    int rowA = waveId * 16 + (lane & 15);
    if (rowA >= M) rowA = M - 1;         // clamp: garbage rows never stored
    int kPair = (lane >> 4) * 2;
    int nLane = lane & 15;

    // clamped per-tile column indices (no divergence, no OOB)
    int colIdx[TILES_N];
#pragma unroll
    for (int t = 0; t < TILES_N; ++t) {
        int c = t * 16 + nLane;
        colIdx[t] = (c < NOUT) ? c : (NOUT - 1);
    }

    v8f acc[TILES_N] = {};

    const float* ap = A + (size_t)rowA * K + kPair;   // 8B-aligned (kPair even)
    const float* bp = B + (size_t)kPair * NOUT;

#pragma unroll
    for (int k0 = 0; k0 < K; k0 += 4) {
        float2 av = *(const float2*)ap;               // global_load_b64
        v2f a;
        a[0] = av.x;
        a[1] = av.y;
#pragma unroll
        for (int t = 0; t < TILES_N; ++t) {
            v2f b;
            b[0] = bp[colIdx[t]];
            b[1] = bp[NOUT + colIdx[t]];
            acc[t] = __builtin_amdgcn_wmma_f32_16x16x4_f32(
                false, a, false, b, (short)0, acc[t], false, false);
        }
        ap += 4;
        bp += 4 * NOUT;
    }

    int rBase = waveId * 16 + (lane >> 4) * 8;
#pragma unroll
    for (int t = 0; t < TILES_N; ++t) {
        int c = t * 16 + nLane;
        if (c < NOUT) {                               // compile-time true for NOUT=128
            float bv = bias ? bias[c] : 0.0f;
#pragma unroll
            for (int v = 0; v < 8; ++v) {
                int r = rBase + v;
                if (r < M) C[(size_t)r * NOUT + c] = acc[t][v] + bv;
            }
        }
    }
}

// ---------------------------------------------------------------------------
// h_new[n][:] = bc[:] + hw[n][:] * dinv[n]^2   (bias + self-loop, no atomics)
// one wave per node, float4 per lane (32*4 = 128)
// ---------------------------------------------------------------------------
__global__ void k_self_init(const float* __restrict__ hw, const float* __restrict__ dinv,
                            const float* __restrict__ bc, float* __restrict__ hnew, int n) {
    int idx  = blockIdx.x * blockDim.x + threadIdx.x;
    int node = idx >> 5;
    int lane = idx & 31;
    if (node >= n) return;
    float w = dinv[node];
    w *= w;
    float4 v  = ((const float4*)(hw + (size_t)node * D_HID))[lane];
    float4 bb = ((const float4*)bc)[lane];
    float4 o;
    o.x = bb.x + v.x * w;
    o.y = bb.y + v.y * w;
    o.z = bb.z + v.z * w;
    o.w = bb.w + v.w * w;
    ((float4*)(hnew + (size_t)node * D_HID))[lane] = o;
}

// ---------------------------------------------------------------------------
// Edge scatter: h_new[dst] += hw[src] * dinv[src]*dinv[dst]
// one wave per edge: 512B gather + 4 float atomics per lane (L2-resident)
// ---------------------------------------------------------------------------
__global__ void k_edge_scatter(const float* __restrict__ hw,
                               const int* __restrict__ src, const int* __restrict__ dst,
                               const float* __restrict__ dinv, float* __restrict__ hnew, int e) {
    int idx  = blockIdx.x * blockDim.x + threadIdx.x;
    int edge = idx >> 5;
    int lane = idx & 31;
    if (edge >= e) return;
    int s = src[edge];
    int d = dst[edge];
    float w = dinv[s] * dinv[d];
    float4 v = ((const float4*)(hw + (size_t)s * D_HID))[lane];
    float* o = hnew + (size_t)d * D_HID + lane * 4;
    unsafeAtomicAdd(o + 0, v.x * w);
    unsafeAtomicAdd(o + 1, v.y * w);
    unsafeAtomicAdd(o + 2, v.z * w);
    unsafeAtomicAdd(o + 3, v.w * w);
}

// ---------------------------------------------------------------------------
// LayerNorm(128) + ReLU in place, one wave32 per node, shfl_xor reductions
// ---------------------------------------------------------------------------
__global__ void k_ln_relu(float* __restrict__ h, const float* __restrict__ g,
                          const float* __restrict__ b, int n) {
    int idx  = blockIdx.x * blockDim.x + threadIdx.x;
    int node = idx >> 5;
    int lane = idx & 31;
    if (node >= n) return;
    float4 v = ((const float4*)(h + (size_t)node * D_HID))[lane];
    float s  = v.x + v.y + v.z + v.w;
    float ss = v.x * v.x + v.y * v.y + v.z * v.z + v.w * v.w;
#pragma unroll
    for (int off = 16; off > 0; off >>= 1) {
        s  += __shfl_xor(s, off, 32);
        ss += __shfl_xor(ss, off, 32);
    }
    float mu  = s * (1.0f / 128.0f);
    float var = ss * (1.0f / 128.0f) - mu * mu;
    float inv = rsqrtf(var + EPS_LN);
    float4 gv = ((const float4*)g)[lane];
    float4 bv = ((const float4*)b)[lane];
    float4 o;
    o.x = fmaxf((v.x - mu) * inv * gv.x + bv.x, 0.0f);
    o.y = fmaxf((v.y - mu) * inv * gv.y + bv.y, 0.0f);
    o.z = fmaxf((v.z - mu) * inv * gv.z + bv.z, 0.0f);
    o.w = fmaxf((v.w - mu) * inv * gv.w + bv.w, 0.0f);
    ((float4*)(h + (size_t)node * D_HID))[lane] = o;
}

// ---------------------------------------------------------------------------
// Driver
// ---------------------------------------------------------------------------
static inline int divUp(int a, int b) { return (a + b - 1) / b; }

extern "C" void kernel_launch(void* const* d_in, const int* in_sizes, int n_in,
                              void* d_out, int out_size, void* d_ws, size_t ws_size,
                              hipStream_t stream) {
    const float* x      = (const float*)d_in[0];
    const int*   ei     = (const int*)d_in[1];
    const float* W_enc  = (const float*)d_in[2];
    const float* b_enc  = (const float*)d_in[3];
    const float* W_pre  = (const float*)d_in[4];
    const float* b_pre  = (const float*)d_in[5];
    const float* Wc     = (const float*)d_in[6];
    const float* bc     = (const float*)d_in[7];
    const float* ln_g   = (const float*)d_in[8];
    const float* ln_b   = (const float*)d_in[9];
    const float* W_head = (const float*)d_in[10];
    const float* b_head = (const float*)d_in[11];

    const int DIN  = 100;
    const int D    = D_HID;
    const int L    = 3;
    const int N    = in_sizes[0] / DIN;
    const int E    = in_sizes[1] / 2;

    float* bufA = (float*)d_ws;                 // h          [N, 128]
    float* bufB = bufA + (size_t)N * D;         // hw / tmp   [N, 128]
    float* dinv = bufB + (size_t)N * D;         // deg->dinv  [N]

    const int* srcIdx = ei;      // edge_index[0]
    const int* dstIdx = ei + E;  // edge_index[1]

    // --- degree & symmetric normalization ---
    k_fill1<<<divUp(N, 256), 256, 0, stream>>>(dinv, N);
    k_deg_accum<<<divUp(E, 256), 256, 0, stream>>>(dstIdx, dinv, E);
    k_rsqrt_inplace<<<divUp(N, 256), 256, 0, stream>>>(dinv, N);

    // one wave per 16-row block (8 waves / 256-thread block)
    int rowBlocks = divUp(N, 16);
    int gemmGrid  = divUp(rowBlocks, 8);

    // --- encoder + pre-MP linear ---
    k_gemm_f32_wmma<100, 128><<<gemmGrid, 256, 0, stream>>>(x, W_enc, b_enc, bufB, N);
    k_gemm_f32_wmma<128, 128><<<gemmGrid, 256, 0, stream>>>(bufB, W_pre, b_pre, bufA, N);

    // --- GCN layers ---
    for (int l = 0; l < L; ++l) {
        k_gemm_f32_wmma<128, 128><<<gemmGrid, 256, 0, stream>>>(
            bufA, Wc + (size_t)l * D * D, nullptr, bufB, N);
        k_self_init<<<divUp(N * 32, 256), 256, 0, stream>>>(bufB, dinv, bc + l * D, bufA, N);
        k_edge_scatter<<<divUp(E * 32, 256), 256, 0, stream>>>(bufB, srcIdx, dstIdx, dinv, bufA, E);
        if (l < L - 1)
            k_ln_relu<<<divUp(N * 32, 256), 256, 0, stream>>>(bufA, ln_g + l * D, ln_b + l * D, N);
    }

    // --- classification head ---
    k_gemm_f32_wmma<128, 40><<<gemmGrid, 256, 0, stream>>>(bufA, W_head, b_head, (float*)d_out, N);
}